// GCN_32_6064493821953
// MI455X (gfx1250) — compile-verified
//
#include <hip/hip_runtime.h>
#include <math.h>

// Problem constants (fixed by the reference).
#define N_NODES 100000
#define N_EDGES 1600000
#define FEAT 32
#define HID 32
#define NCLS 10
#define NGRAPH 64

typedef __attribute__((ext_vector_type(2))) float v2f;
typedef __attribute__((ext_vector_type(8))) float v8f;

// ---------------- init: degrees, best/prop/valid, graph accumulators -------
__global__ void k_init(float* degf, float* dgf, unsigned* best, int* prop,
                       float* valid, float* gsum, float* gcnt) {
  int i = blockIdx.x * blockDim.x + threadIdx.x;
  if (i < N_NODES) {
    degf[i] = 1.0f;   // self-loop contributes 1 to dst-degree
    dgf[i]  = 0.0f;
    best[i] = 0u;     // bits of +0.0f; all cut values are > 0
    prop[i] = -1;
    valid[i] = 0.0f;
  }
  if (i < NGRAPH * HID) gsum[i] = 0.0f;
  if (i < NGRAPH) gcnt[i] = 0.0f;
}

// ---------------- degree counting ------------------------------------------
__global__ void k_count(const int* __restrict__ src, const int* __restrict__ dst,
                        float* degf, float* dgf) {
  int e = blockIdx.x * blockDim.x + threadIdx.x;
  if (e >= N_EDGES) return;
  atomicAdd(&degf[dst[e]], 1.0f);  // in-degree (+1 self-loop already)
  atomicAdd(&dgf[src[e]], 1.0f);   // out-degree (no self-loops)
}

__global__ void k_dinv(const float* __restrict__ degf, float* dinv) {
  int i = blockIdx.x * blockDim.x + threadIdx.x;
  if (i >= N_NODES) return;
  dinv[i] = 1.0f / sqrtf(degf[i]);
}

// ---------------- h = x @ W1 via V_WMMA_F32_16X16X4_F32 --------------------
// One wave -> 16 rows x 32 cols. 6250 waves total (N divides exactly).
__global__ void __launch_bounds__(256) k_gemm(const float* __restrict__ x,
                                              const float* __restrict__ W1,
                                              float* __restrict__ h) {
  int lane = threadIdx.x & 31;
  int wave = blockIdx.x * (blockDim.x >> 5) + (threadIdx.x >> 5);
  int row0 = wave * 16;
  if (row0 >= N_NODES) return;  // wave-uniform: EXEC stays all-1s for WMMA

  int m    = lane & 15;          // row (A) / col (B,D) within tile
  int koff = (lane >> 4) * 2;    // low lanes: K=0,1 ; high lanes: K=2,3

  // Preload all B fragments: 8 K-steps x 2 N-tiles.
  v2f bf[2][8];
#pragma unroll
  for (int k = 0; k < 8; ++k) {
    int kr = k * 4 + koff;
#pragma unroll
    for (int nt = 0; nt < 2; ++nt) {
      int col = nt * 16 + m;
      v2f b;
      b.x = W1[kr * HID + col];
      b.y = W1[(kr + 1) * HID + col];
      bf[nt][k] = b;
    }
  }

  v8f acc0 = {0.f, 0.f, 0.f, 0.f, 0.f, 0.f, 0.f, 0.f};
  v8f acc1 = {0.f, 0.f, 0.f, 0.f, 0.f, 0.f, 0.f, 0.f};

  const float* xr = x + (size_t)(row0 + m) * FEAT + koff;
#pragma unroll
  for (int k = 0; k < 8; ++k) {
    v2f a = *(const v2f*)(xr + k * 4);  // 8B-aligned
    acc0 = __builtin_amdgcn_wmma_f32_16x16x4_f32(false, a, false, bf[0][k],
                                                 (short)0, acc0, false, false);
    acc1 = __builtin_amdgcn_wmma_f32_16x16x4_f32(false, a, false, bf[1][k],
                                                 (short)0, acc1, false, false);
  }

  int rbase = row0 + (lane >> 4) * 8;  // D: VGPR v -> row v (low) / v+8 (high)
#pragma unroll
  for (int v = 0; v < 8; ++v) {
    h[(size_t)(rbase + v) * HID + m]      = acc0[v];
    h[(size_t)(rbase + v) * HID + 16 + m] = acc1[v];
  }
}

// ---------------- agg = b1 + dinv[i]^2 * h[i]  (self-loop term) ------------
__global__ void k_agginit(const float* __restrict__ h, const float* __restrict__ dinv,
                          const float* __restrict__ b1, float* agg) {
  int i = blockIdx.x * 8 + (threadIdx.x >> 5);
  int j = threadIdx.x & 31;
  if (i >= N_NODES) return;
  float dv = dinv[i];
  agg[(size_t)i * HID + j] = b1[j] + dv * dv * h[(size_t)i * HID + j];
}

// ---------------- edge scatter: agg[d] += dinv[s]*dinv[d]*h[s] -------------
__global__ void k_scatter(const int* __restrict__ src, const int* __restrict__ dst,
                          const float* __restrict__ dinv, const float* __restrict__ h,
                          float* agg) {
  int e = blockIdx.x * 8 + (threadIdx.x >> 5);
  int j = threadIdx.x & 31;
  if (e >= N_EDGES) return;
  int s = src[e], d = dst[e];
  float nrm = dinv[s] * dinv[d];
  atomicAdd(&agg[(size_t)d * HID + j], nrm * h[(size_t)s * HID + j]);
}

// ---------------- relu in place; zero pooled buffer (reuses h) -------------
__global__ void k_relu_zero(float* agg, float* pooled) {
  size_t t = (size_t)blockIdx.x * blockDim.x + threadIdx.x;
  if (t >= (size_t)N_NODES * HID) return;
  agg[t] = fmaxf(agg[t], 0.0f);
  pooled[t] = 0.0f;  // bits == 0 for unsigned atomicMax
}

// ---------------- normalized cut + segment_max(cut, src) -------------------
__global__ void k_cutbest(const int* __restrict__ src, const int* __restrict__ dst,
                          const float* __restrict__ pos, const float* __restrict__ dgf,
                          float* cut, unsigned* best) {
  int e = blockIdx.x * blockDim.x + threadIdx.x;
  if (e >= N_EDGES) return;
  int s = src[e], d = dst[e];
  float dx = pos[2 * s] - pos[2 * d];
  float dy = pos[2 * s + 1] - pos[2 * d + 1];
  float w = sqrtf(dx * dx + dy * dy + 1e-12f);
  float c = w * (1.0f / fmaxf(dgf[s], 1.0f) + 1.0f / fmaxf(dgf[d], 1.0f));
  cut[e] = c;
  atomicMax(&best[s], __float_as_uint(c));  // c > 0, bit order == float order
}

// ---------------- prop = segment_max(cand, src) ----------------------------
__global__ void k_prop(const int* __restrict__ src, const int* __restrict__ dst,
                       const float* __restrict__ cut, const unsigned* __restrict__ best,
                       int* prop) {
  int e = blockIdx.x * blockDim.x + threadIdx.x;
  if (e >= N_EDGES) return;
  int s = src[e];
  int cand = (cut[e] >= __uint_as_float(best[s])) ? dst[e] : -1;
  atomicMax(&prop[s], cand);
}

// ---------------- graclus mutual matching ----------------------------------
__global__ void k_cluster(const int* __restrict__ prop, int* cluster) {
  int i = blockIdx.x * blockDim.x + threadIdx.x;
  if (i >= N_NODES) return;
  int p = prop[i];
  int ps = min(max(p, 0), N_NODES - 1);
  bool mutual = (p >= 0) && (p != i) && (prop[ps] == i);
  cluster[i] = mutual ? min(i, ps) : i;
}

// ---------------- pooled = segment_max(h1, cluster); valid flags -----------
__global__ void k_pool(const int* __restrict__ cluster, const float* __restrict__ h1,
                       unsigned* pooled_u, float* valid) {
  int i = blockIdx.x * 8 + (threadIdx.x >> 5);
  int j = threadIdx.x & 31;
  if (i >= N_NODES) return;
  int c = cluster[i];
  atomicMax(&pooled_u[(size_t)c * HID + j],
            __float_as_uint(h1[(size_t)i * HID + j]));  // h1 >= 0 after relu
  if (j == 0) valid[c] = 1.0f;
}

// ---------------- per-graph sums -------------------------------------------
__global__ void k_gpool(const int* __restrict__ batch, const float* __restrict__ pooled,
                        const float* __restrict__ valid, float* gsum, float* gcnt) {
  int i = blockIdx.x * 8 + (threadIdx.x >> 5);
  int j = threadIdx.x & 31;
  if (i >= N_NODES) return;
  int b = batch[i];
  atomicAdd(&gsum[b * HID + j], pooled[(size_t)i * HID + j]);
  if (j == 0) atomicAdd(&gcnt[b], valid[i]);
}

// ---------------- mean -> fc -> log_softmax --------------------------------
__global__ void k_final(const float* __restrict__ gsum, const float* __restrict__ gcnt,
                        const float* __restrict__ fc_w, const float* __restrict__ fc_b,
                        float* out) {
  int b = threadIdx.x;
  if (b >= NGRAPH) return;
  float inv = 1.0f / fmaxf(gcnt[b], 1.0f);
  float mean[HID];
#pragma unroll
  for (int j = 0; j < HID; ++j) mean[j] = gsum[b * HID + j] * inv;
  float logits[NCLS];
  float mx = -INFINITY;
#pragma unroll
  for (int c = 0; c < NCLS; ++c) {
    float a = fc_b[c];
#pragma unroll
    for (int j = 0; j < HID; ++j) a += mean[j] * fc_w[j * NCLS + c];
    logits[c] = a;
    mx = fmaxf(mx, a);
  }
  float s = 0.0f;
#pragma unroll
  for (int c = 0; c < NCLS; ++c) s += expf(logits[c] - mx);
  float lse = mx + logf(s);
#pragma unroll
  for (int c = 0; c < NCLS; ++c) out[b * NCLS + c] = logits[c] - lse;
}

extern "C" void kernel_launch(void* const* d_in, const int* in_sizes, int n_in,
                              void* d_out, int out_size, void* d_ws, size_t ws_size,
                              hipStream_t stream) {
  const float* x    = (const float*)d_in[0];
  const float* pos  = (const float*)d_in[1];
  const float* W1   = (const float*)d_in[2];
  const float* b1   = (const float*)d_in[3];
  const float* fc_w = (const float*)d_in[4];
  const float* fc_b = (const float*)d_in[5];
  const int*   ei   = (const int*)d_in[6];
  const int*   batch= (const int*)d_in[7];
  const int* src = ei;
  const int* dst = ei + N_EDGES;
  float* out = (float*)d_out;

  // Workspace carve-up (256B aligned blocks).
  char* p = (char*)d_ws;
  auto take = [&](size_t bytes) { char* r = p; p += (bytes + 255) & ~(size_t)255; return r; };
  float*    h      = (float*)take((size_t)N_NODES * HID * 4);  // reused as pooled
  float*    agg    = (float*)take((size_t)N_NODES * HID * 4);
  float*    cut    = (float*)take((size_t)N_EDGES * 4);
  float*    degf   = (float*)take((size_t)N_NODES * 4);
  float*    dgf    = (float*)take((size_t)N_NODES * 4);
  float*    dinv   = (float*)take((size_t)N_NODES * 4);
  unsigned* best   = (unsigned*)take((size_t)N_NODES * 4);
  int*      prop   = (int*)take((size_t)N_NODES * 4);
  int*      clus   = (int*)take((size_t)N_NODES * 4);
  float*    valid  = (float*)take((size_t)N_NODES * 4);
  float*    gsum   = (float*)take((size_t)NGRAPH * HID * 4);
  float*    gcnt   = (float*)take((size_t)NGRAPH * 4);
  float*    pooled = h;  // h is dead after k_scatter

  const int T = 256;
  int gN   = (N_NODES + T - 1) / T;               // 391
  int gE   = (N_EDGES + T - 1) / T;               // 6250
  int gN32 = (N_NODES * HID + T - 1) / T;         // 12500 (8 rows / block)
  int gE32 = (N_EDGES + 7) / 8;                   // 200000 (8 edges / block)
  int gW   = ((N_NODES / 16) + 7) / 8;            // 782 blocks of 8 waves

  k_init     <<<gN,   T, 0, stream>>>(degf, dgf, best, prop, valid, gsum, gcnt);
  k_count    <<<gE,   T, 0, stream>>>(src, dst, degf, dgf);
  k_dinv     <<<gN,   T, 0, stream>>>(degf, dinv);
  k_gemm     <<<gW,   T, 0, stream>>>(x, W1, h);
  k_agginit  <<<gN32, T, 0, stream>>>(h, dinv, b1, agg);
  k_scatter  <<<gE32, T, 0, stream>>>(src, dst, dinv, h, agg);
  k_relu_zero<<<gN32, T, 0, stream>>>(agg, pooled);
  k_cutbest  <<<gE,   T, 0, stream>>>(src, dst, pos, dgf, cut, best);
  k_prop     <<<gE,   T, 0, stream>>>(src, dst, cut, best, prop);
  k_cluster  <<<gN,   T, 0, stream>>>(prop, clus);
  k_pool     <<<gN32, T, 0, stream>>>(clus, agg, (unsigned*)pooled, valid);
  k_gpool    <<<gN32, T, 0, stream>>>(batch, pooled, valid, gsum, gcnt);
  k_final    <<<1,   64, 0, stream>>>(gsum, gcnt, fc_w, fc_b, out);
}